// Model_90357521973849
// MI455X (gfx1250) — compile-verified
//
#include <hip/hip_runtime.h>
#include <math.h>

typedef __attribute__((ext_vector_type(2))) float v2f;
typedef __attribute__((ext_vector_type(8))) float v8f;

#define BB 32
#define MM 64
#define LL 512
#define TT 96
#define DD 64
#define NN 128

// ---------------------------------------------------------------------------
// Embed: x[B,M,L] --(edge-pad 4, conv k=8 s=4, 1->64ch)--> xemb[B,M,D,N]
// gid = ((b*M+m)*D + d)*N + n ; exact grid, no guard.
// ---------------------------------------------------------------------------
__global__ void embed_kernel(const float* __restrict__ x, const float* __restrict__ ew,
                             const float* __restrict__ eb, float* __restrict__ xemb) {
  size_t gid = (size_t)blockIdx.x * blockDim.x + threadIdx.x;
  int n = (int)(gid & 127);
  int d = (int)((gid >> 7) & 63);
  size_t bm = gid >> 13;
  const float* xr = x + bm * LL;
  float acc = eb[d];
#pragma unroll
  for (int p = 0; p < 8; ++p) {
    int i = n * 4 + p;
    i = (i > LL - 1) ? (LL - 1) : i;          // replicate (edge) pad on the right
    acc = fmaf(xr[i], ew[d * 8 + p], acc);
  }
  xemb[gid] = acc;
}

// ---------------------------------------------------------------------------
// Depthwise conv over M (k=4, pad left 1 / right 2, groups=N*D), fused with
// the [B,M,D,N] -> [B,D,M,N] transpose.  gid = ((b*D+d)*M+m)*N + n.
// ---------------------------------------------------------------------------
__global__ void dwconv_kernel(const float* __restrict__ xemb, const float* __restrict__ w,
                              const float* __restrict__ bias, float* __restrict__ out) {
  size_t gid = (size_t)blockIdx.x * blockDim.x + threadIdx.x;
  int n = (int)(gid & 127);
  int m = (int)((gid >> 7) & 63);
  int d = (int)((gid >> 13) & 63);
  int b = (int)(gid >> 19);
  int c = n * DD + d;                          // channel index in reference layout
  float acc = bias[c];
#pragma unroll
  for (int k = 0; k < 4; ++k) {
    int mm = m - 1 + k;                        // 'same' pad: left 1, right 2
    if (mm >= 0 && mm < MM)
      acc = fmaf(xemb[(((size_t)b * MM + mm) * DD + d) * NN + n], w[c * 4 + k], acc);
  }
  out[gid] = acc;
}

// ---------------------------------------------------------------------------
// Rearrange a [rows, K] weight matrix into per-wave WMMA A-fragment order:
// wf[ct][kc][lane] = float2{ W[ct*16+lane%16, kc*4+koff], W[..., +1] }
// with koff = 0 for lanes 0-15, 2 for lanes 16-31 (ISA 16x4 fp32 A layout).
// ---------------------------------------------------------------------------
__global__ void prep_wf_kernel(const float* __restrict__ W, v2f* __restrict__ wf,
                               int K, int nct) {
  int KC = K >> 2;
  int total = nct * KC * 32;
  for (int i = blockIdx.x * blockDim.x + threadIdx.x; i < total;
       i += gridDim.x * blockDim.x) {
    int lane = i & 31;
    int t = i >> 5;
    int kc = t % KC;
    int ct = t / KC;
    int row = ct * 16 + (lane & 15);
    int k = kc * 4 + ((lane < 16) ? 0 : 2);
    v2f v;
    v.x = W[(size_t)row * K + k];
    v.y = W[(size_t)row * K + k + 1];
    wf[i] = v;
  }
}

// ---------------------------------------------------------------------------
// Implicit-GEMM conv2d (4x3 taps, pad (1,2)/(1,1)) via V_WMMA_F32_16X16X4_F32.
// One wave = 16 co x 16 n tile at fixed (b,m).  K = ci*12 + kh*3 + kw.
// A = weights (pre-arranged fragment stream, coalesced b64/lane).
// B = activations from LDS halo tile  [CIN][4 rows][20 cols].
// DO_GELU path writes H[B,COUT,M,N]; else fuses residual in place into
// xemb[B,M,D,N].
// ---------------------------------------------------------------------------
template <int CIN, int COUT, bool DO_GELU>
__global__ __launch_bounds__(COUT * 2) void conv_ffn_kernel(
    const float* __restrict__ Ain, const v2f* __restrict__ wf,
    const float* __restrict__ bias, float* __restrict__ Out) {
  __shared__ float lds[CIN * 80];
  const int tid = threadIdx.x;
  const int nt = blockIdx.x & 7;               // N/16 tiles
  const int m  = (blockIdx.x >> 3) & 63;
  const int b  = blockIdx.x >> 9;
  const int n0 = nt * 16;

  // Cooperative halo load (zero pad outside [0,M) x [0,N)).
  for (int idx = tid; idx < CIN * 72; idx += COUT * 2) {
    int ci  = idx / 72;
    int rem = idx - ci * 72;
    int r   = rem / 18;
    int col = rem - r * 18;
    int mm  = m - 1 + r;
    int nn  = n0 - 1 + col;
    float v = 0.0f;
    if (mm >= 0 && mm < MM && (unsigned)nn < (unsigned)NN)
      v = Ain[(((size_t)b * CIN + ci) * MM + mm) * NN + nn];
    lds[ci * 80 + r * 20 + col] = v;
  }
  __syncthreads();

  const int lane  = tid & 31;
  const int wt    = tid >> 5;                  // co tile owned by this wave
  const int nloc  = lane & 15;
  const int koff0 = (lane < 16) ? 0 : 2;
  const int hi    = (lane < 16) ? 0 : 8;
  const int co0   = wt * 16;

  // Per-lane LDS offsets for the 3 K-chunks inside one ci (12 taps).
  int offA[3], offB[3];
#pragma unroll
  for (int rc = 0; rc < 3; ++rc) {
    int t0 = rc * 4 + koff0;
    int t1 = t0 + 1;
    offA[rc] = (t0 / 3) * 20 + (t0 % 3) + nloc;
    offB[rc] = (t1 / 3) * 20 + (t1 % 3) + nloc;
  }

  v8f acc;
#pragma unroll
  for (int r = 0; r < 8; ++r) acc[r] = bias[co0 + r + hi];

  const v2f* wp = wf + (size_t)wt * (CIN * 3) * 32 + lane;
  for (int ci = 0; ci < CIN; ++ci) {
    const float* lb = lds + ci * 80;
    __builtin_prefetch(wp + 32 * 6, 0, 0);     // global_prefetch_b8 on weight stream
#pragma unroll
    for (int rc = 0; rc < 3; ++rc) {
      v2f bv;
      bv.x = lb[offA[rc]];
      bv.y = lb[offB[rc]];
      v2f av = *wp;
      wp += 32;
      acc = __builtin_amdgcn_wmma_f32_16x16x4_f32(false, av, false, bv,
                                                  (short)0, acc, false, false);
    }
  }

  const int n = n0 + nloc;
  if (DO_GELU) {
#pragma unroll
    for (int r = 0; r < 8; ++r) {
      int co = co0 + r + hi;
      float v = acc[r];
      v = 0.5f * v * (1.0f + erff(v * 0.70710678118654752f));  // exact GELU
      Out[(((size_t)b * COUT + co) * MM + m) * NN + n] = v;
    }
  } else {
    // Out == xemb [B,M,D,N]; residual is element-wise -> safe in place.
#pragma unroll
    for (int r = 0; r < 8; ++r) {
      int co = co0 + r + hi;
      size_t oi = (((size_t)b * MM + m) * DD + co) * NN + n;
      Out[oi] = acc[r] + Out[oi];
    }
  }
}

// ---------------------------------------------------------------------------
// Head GEMM: pred[bm, t] = z[bm, :8192] . head_w[t, :8192] + head_b[t]
// Wave tile: 16 t x 16 bm ; A = head_w fragments, B = z rows (fp32 WMMA).
// grid = 128 (bm tiles), block = 192 (6 waves = all t tiles).
// ---------------------------------------------------------------------------
__global__ __launch_bounds__(192) void head_kernel(
    const float* __restrict__ z, const v2f* __restrict__ wfh,
    const float* __restrict__ hb, float* __restrict__ out) {
  const int tid   = threadIdx.x;
  const int lane  = tid & 31;
  const int wt    = tid >> 5;                  // 0..5 -> t tile
  const int bm0   = blockIdx.x * 16;
  const int nloc  = lane & 15;
  const int koff0 = (lane < 16) ? 0 : 2;
  const int hi    = (lane < 16) ? 0 : 8;
  const int t0    = wt * 16;

  v8f acc;
#pragma unroll
  for (int r = 0; r < 8; ++r) acc[r] = hb[t0 + r + hi];

  const v2f* wp = wfh + (size_t)wt * 2048 * 32 + lane;
  const float* zr = z + (size_t)(bm0 + nloc) * 8192 + koff0;
  for (int kc = 0; kc < 2048; ++kc) {
    v2f bv = *(const v2f*)(zr + kc * 4);       // 8B aligned: even float offset
    v2f av = *wp;
    wp += 32;
    acc = __builtin_amdgcn_wmma_f32_16x16x4_f32(false, av, false, bv,
                                                (short)0, acc, false, false);
  }
#pragma unroll
  for (int r = 0; r < 8; ++r)
    out[(size_t)(bm0 + nloc) * TT + t0 + r + hi] = acc[r];
}

// ---------------------------------------------------------------------------
extern "C" void kernel_launch(void* const* d_in, const int* in_sizes, int n_in,
                              void* d_out, int out_size, void* d_ws, size_t ws_size,
                              hipStream_t stream) {
  const float* x      = (const float*)d_in[0];
  const float* emb_w  = (const float*)d_in[1];
  const float* emb_b  = (const float*)d_in[2];
  const float* head_w = (const float*)d_in[3];
  const float* head_b = (const float*)d_in[4];
  const float* dw_w[2] = {(const float*)d_in[5],  (const float*)d_in[11]};
  const float* dw_b[2] = {(const float*)d_in[6],  (const float*)d_in[12]};
  const float* f1_w[2] = {(const float*)d_in[7],  (const float*)d_in[13]};
  const float* f1_b[2] = {(const float*)d_in[8],  (const float*)d_in[14]};
  const float* f2_w[2] = {(const float*)d_in[9],  (const float*)d_in[15]};
  const float* f2_b[2] = {(const float*)d_in[10], (const float*)d_in[16]};

  float* ws   = (float*)d_ws;
  float* xemb = ws;                                   // 33,554,432 f32 [B,M,D,N]
  float* Abuf = xemb + (size_t)33554432;              // 33,554,432 f32 [B,D,M,N]
  float* Hbuf = Abuf + (size_t)33554432;              // 67,108,864 f32 [B,2D,M,N]
  v2f*  wf1   = (v2f*)(Hbuf + (size_t)67108864);      //  49,152 v2f
  v2f*  wf2   = wf1 + (size_t)49152;                  //  49,152 v2f
  v2f*  wfh   = wf2 + (size_t)49152;                  // 393,216 v2f

  const int elem_total = BB * MM * DD * NN;           // 16,777,216
  embed_kernel<<<elem_total / 256, 256, 0, stream>>>(x, emb_w, emb_b, xemb);

  for (int blk = 0; blk < 2; ++blk) {
    dwconv_kernel<<<elem_total / 256, 256, 0, stream>>>(xemb, dw_w[blk], dw_b[blk], Abuf);

    prep_wf_kernel<<<192, 256, 0, stream>>>(f1_w[blk], wf1, /*K=*/DD * 12, /*nct=*/8);
    conv_ffn_kernel<DD, 2 * DD, true>
        <<<BB * MM * (NN / 16), 2 * DD * 2, 0, stream>>>(Abuf, wf1, f1_b[blk], Hbuf);

    prep_wf_kernel<<<192, 256, 0, stream>>>(f2_w[blk], wf2, /*K=*/2 * DD * 12, /*nct=*/4);
    conv_ffn_kernel<2 * DD, DD, false>
        <<<BB * MM * (NN / 16), DD * 2, 0, stream>>>(Hbuf, wf2, f2_b[blk], xemb);
  }

  prep_wf_kernel<<<1536, 256, 0, stream>>>(head_w, wfh, /*K=*/DD * NN, /*nct=*/TT / 16);
  head_kernel<<<(BB * MM) / 16, 192, 0, stream>>>(xemb, wfh, head_b, (float*)d_out);
}